// InputAttentionHead_39960375722666
// MI455X (gfx1250) — compile-verified
//
#include <hip/hip_runtime.h>
#include <math.h>

typedef float v2f __attribute__((ext_vector_type(2)));
typedef float v8f __attribute__((ext_vector_type(8)));

// ---------------------------------------------------------------------------
// Kernel 1: sinusoidal positional-encoding table, 528 rows x 512 cols (rows
// padded 513 -> 528 so downstream WMMA tiles need no bounds checks).
// pe[n, 2i] = sin(n * exp(-2i*ln(1e4)/512)), pe[n, 2i+1] = cos(...)
// ---------------------------------------------------------------------------
__global__ void ia_pe_table_kernel(float* __restrict__ pe) {
    const int n = blockIdx.x;       // 0..527
    const int i = threadIdx.x;      // 0..255 (pair index)
    const float div = __expf(-(float)(2 * i) * 0.01798894604f); // ln(10000)/512
    const float ang = (float)n * div;
    float s, c;
    __sincosf(ang, &s, &c);
    pe[(size_t)n * 512 + 2 * i]     = s;
    pe[(size_t)n * 512 + 2 * i + 1] = c;
}

// ---------------------------------------------------------------------------
// WMMA f32 GEMM: C[M,N] = A[M,K] @ B (BT=false: B is KxN; BT=true: B is NxK
// in memory, i.e. we multiply by B^T). One 16x16 tile per wave, 4 waves per
// block (16 rows x 64 cols per block). M, 16 | N tile-guarded, K % 4 == 0.
// Uses V_WMMA_F32_16X16X4_F32 (exact fp32 accumulate path on CDNA5).
// ---------------------------------------------------------------------------
template <bool BT>
__global__ __launch_bounds__(128)
void ia_wmma_gemm_f32(const float* __restrict__ A,
                      const float* __restrict__ B,
                      float* __restrict__ C,
                      int N, int K, int lda, int ldb, int ldc) {
    const int lane = threadIdx.x & 31;
    const int wave = threadIdx.x >> 5;
    const int row0 = blockIdx.y * 16;
    const int col0 = blockIdx.x * 64 + wave * 16;
    if (col0 >= N) return;                 // tail tiles (N multiple of 16)

    const int mn   = lane & 15;            // M index for A, N index for B
    const int koff = (lane >> 4) << 1;     // lanes 0-15 -> K+0/1, 16-31 -> K+2/3

    v8f acc = {0.f, 0.f, 0.f, 0.f, 0.f, 0.f, 0.f, 0.f};
    const float* arow = A + (size_t)(row0 + mn) * lda + koff;

    for (int k = 0; k < K; k += 4) {
        v2f a, b;
        {   // A fragment: row (row0+mn), cols k+koff, k+koff+1 (8B load)
            const float2 t = *(const float2*)(arow + k);
            a.x = t.x; a.y = t.y;
        }
        if (BT) {
            const float2 t = *(const float2*)(B + (size_t)(col0 + mn) * ldb + k + koff);
            b.x = t.x; b.y = t.y;
        } else {
            b.x = B[(size_t)(k + koff)     * ldb + col0 + mn];
            b.y = B[(size_t)(k + koff + 1) * ldb + col0 + mn];
        }
        acc = __builtin_amdgcn_wmma_f32_16x16x4_f32(
            /*neg_a=*/false, a, /*neg_b=*/false, b,
            /*c_mod=*/(short)0, acc, /*reuse_a=*/false, /*reuse_b=*/false);
    }

    // C/D layout: VGPR j holds M = j (lanes 0-15) or j+8 (lanes 16-31), N = lane&15
    const int rbase = row0 + ((lane < 16) ? 0 : 8);
    float* crow = C + (size_t)rbase * ldc + col0 + mn;
#pragma unroll
    for (int j = 0; j < 8; ++j) crow[(size_t)j * ldc] = acc[j];
}

// ---------------------------------------------------------------------------
// Kernel 5: fused streaming attention pass. One block (8 waves) per batch b.
// Row n of P: n==0 -> v[b], n>=1 -> phi[b, n-1]. Wave w handles rows
// n = w, w+8, ... < 513 with an independent online softmax; per-lane kq
// fragment lives in registers so the inner loop touches only global memory.
// Outputs attn_w[b, 0..512] and u[b,a] = sum_n attn_w[b,n] * P[b,n,a].
// ---------------------------------------------------------------------------
__global__ __launch_bounds__(256)
void ia_attn_stream_kernel(const float* __restrict__ phi,
                           const float* __restrict__ v,
                           const float* __restrict__ kq,
                           const float* __restrict__ pescore,
                           float* __restrict__ u_out,
                           float* __restrict__ attn_out) {
    const int b    = blockIdx.x;
    const int tid  = threadIdx.x;
    const int lane = tid & 31;
    const int wave = tid >> 5;

    __shared__ float4 up4[8 * 128];    // per-wave partial u (16 KB)
    __shared__ float  sc[516];         // raw scores
    __shared__ float  mlw[8], llw[8];  // per-wave running max / sum

    // lane's kq fragment: elements a = j*128 + lane*4 .. +3 (float4 idx j*32+lane)
    const float4* kq4 = (const float4*)(kq + (size_t)b * 512);
    float4 kr[4];
#pragma unroll
    for (int j = 0; j < 4; ++j) kr[j] = kq4[j * 32 + lane];

    const float*  pesc = pescore + (size_t)b * 528;
    const float4* v4   = (const float4*)(v + (size_t)b * 512);

    float m = -INFINITY, lsum = 0.f;
    float4 ua[4];
#pragma unroll
    for (int j = 0; j < 4; ++j) ua[j] = make_float4(0.f, 0.f, 0.f, 0.f);

    for (int n = wave; n < 513; n += 8) {
        const float4* src = (n == 0)
            ? v4
            : (const float4*)(phi + ((size_t)b * 512 + (n - 1)) * 512);
        if (n + 8 < 513) {   // prefetch this wave's next row (2 KB / 32 lanes)
            const char* pf = (const char*)(phi + ((size_t)b * 512 + (n + 7)) * 512);
            __builtin_prefetch(pf + lane * 64, 0, 3);
        }
        float4 r[4];
#pragma unroll
        for (int j = 0; j < 4; ++j) r[j] = src[j * 32 + lane];

        float dot = 0.f;
#pragma unroll
        for (int j = 0; j < 4; ++j)
            dot += r[j].x * kr[j].x + r[j].y * kr[j].y +
                   r[j].z * kr[j].z + r[j].w * kr[j].w;
#pragma unroll
        for (int off = 16; off > 0; off >>= 1)
            dot += __shfl_xor(dot, off, 32);

        const float s = dot * 0.03125f + pesc[n];   // 1/sqrt(1024)
        if (lane == 0) sc[n] = s;

        const float mnew = fmaxf(m, s);
        const float corr = __expf(m - mnew);        // exp(-inf)=0 on first iter
        const float wgt  = __expf(s - mnew);
        lsum = lsum * corr + wgt;
#pragma unroll
        for (int j = 0; j < 4; ++j) {
            ua[j].x = ua[j].x * corr + wgt * r[j].x;
            ua[j].y = ua[j].y * corr + wgt * r[j].y;
            ua[j].z = ua[j].z * corr + wgt * r[j].z;
            ua[j].w = ua[j].w * corr + wgt * r[j].w;
        }
        m = mnew;
    }

    if (lane == 0) { mlw[wave] = m; llw[wave] = lsum; }
#pragma unroll
    for (int j = 0; j < 4; ++j) up4[wave * 128 + j * 32 + lane] = ua[j];
    __syncthreads();

    // combine the 8 independent online-softmax states
    float M = mlw[0];
#pragma unroll
    for (int w2 = 1; w2 < 8; ++w2) M = fmaxf(M, mlw[w2]);
    float ew[8], L = 0.f;
#pragma unroll
    for (int w2 = 0; w2 < 8; ++w2) { ew[w2] = __expf(mlw[w2] - M); L += llw[w2] * ew[w2]; }
    const float invL = 1.f / L;

    if (tid < 128) {                       // 128 float4 = 512 floats of u
        float4 a = make_float4(0.f, 0.f, 0.f, 0.f);
#pragma unroll
        for (int w2 = 0; w2 < 8; ++w2) {
            const float4 p = up4[w2 * 128 + tid];
            a.x += p.x * ew[w2]; a.y += p.y * ew[w2];
            a.z += p.z * ew[w2]; a.w += p.w * ew[w2];
        }
        a.x *= invL; a.y *= invL; a.z *= invL; a.w *= invL;
        ((float4*)(u_out + (size_t)b * 512))[tid] = a;
    }
    for (int n = tid; n < 513; n += 256)
        attn_out[(size_t)b * 513 + n] = __expf(sc[n] - M) * invL;
}

// ---------------------------------------------------------------------------
// Host launcher
// inputs: h_s(256,1024) phi(256,512,512) v(256,512) wq(1024,1024)
//         wk(512,1024) wv(512,1024)
// outputs: ctx(256,1024) ++ attn_w(256,513)
// ---------------------------------------------------------------------------
extern "C" void kernel_launch(void* const* d_in, const int* in_sizes, int n_in,
                              void* d_out, int out_size, void* d_ws, size_t ws_size,
                              hipStream_t stream) {
    const float* h_s = (const float*)d_in[0];
    const float* phi = (const float*)d_in[1];
    const float* v   = (const float*)d_in[2];
    const float* wq  = (const float*)d_in[3];
    const float* wk  = (const float*)d_in[4];
    const float* wv  = (const float*)d_in[5];

    float* out  = (float*)d_out;
    float* ctx  = out;                 // 256*1024
    float* attn = out + 256 * 1024;    // 256*513

    // workspace layout (floats): ~3.7 MB total
    float* w    = (float*)d_ws;
    float* q    = w;                   // 256*1024 = 262144
    float* kq   = q    + 262144;       // 256*512  = 131072
    float* pe   = kq   + 131072;       // 528*512  = 270336
    float* pesc = pe   + 270336;       // 256*528  = 135168
    float* u    = pesc + 135168;       // 256*512  = 131072

    // 1) PE table (528 rows)
    ia_pe_table_kernel<<<528, 256, 0, stream>>>(pe);

    // 2) q = h_s @ wq           (256x1024, K=1024)
    ia_wmma_gemm_f32<false><<<dim3(16, 16), 128, 0, stream>>>(
        h_s, wq, q, 1024, 1024, 1024, 1024, 1024);

    // 3) kq = q @ wk^T          (256x512, K=1024)
    ia_wmma_gemm_f32<true><<<dim3(8, 16), 128, 0, stream>>>(
        q, wk, kq, 512, 1024, 1024, 1024, 512);

    // 4) pescore = kq @ pe^T    (256x528, K=512)  [cols >=528 tile-guarded]
    ia_wmma_gemm_f32<true><<<dim3(9, 16), 128, 0, stream>>>(
        kq, pe, pesc, 528, 512, 512, 512, 528);

    // 5) fused streaming softmax pass over phi (reads phi exactly once)
    ia_attn_stream_kernel<<<256, 256, 0, stream>>>(phi, v, kq, pesc, u, attn);

    // 6) ctx = u @ wv           (256x1024, K=512)
    ia_wmma_gemm_f32<false><<<dim3(16, 16), 128, 0, stream>>>(
        u, wv, ctx, 1024, 512, 512, 1024, 1024);
}